// GCN_7086696038726
// MI455X (gfx1250) — compile-verified
//
#include <hip/hip_runtime.h>

#define IN_F  128
#define HID_F 64
#define OUT_F 64

typedef __attribute__((ext_vector_type(2))) float v2f;
typedef __attribute__((ext_vector_type(8))) float v8f;

// ---------------- utility: zero a float buffer ----------------
__global__ void gcn_zero_kernel(float* __restrict__ p, long long n) {
    long long i = (long long)blockIdx.x * blockDim.x + threadIdx.x;
    if (i < n) p[i] = 0.0f;
}

// ---------------- degree accumulation ----------------
__global__ void gcn_degree_kernel(const int* __restrict__ src,
                                  const int* __restrict__ dst,
                                  float* __restrict__ deg_out,
                                  float* __restrict__ deg_in, int nE) {
    int e = blockIdx.x * blockDim.x + threadIdx.x;
    if (e < nE) {
        atomicAdd(&deg_out[src[e]], 1.0f);
        atomicAdd(&deg_in [dst[e]], 1.0f);
    }
}

// deg -> rsqrt(max(deg,1)) in place (applied to both arrays contiguously)
__global__ void gcn_norm_kernel(float* __restrict__ p, long long n) {
    long long i = (long long)blockIdx.x * blockDim.x + threadIdx.x;
    if (i < n) {
        float d = p[i];
        d = d < 1.0f ? 1.0f : d;
        p[i] = rsqrtf(d);
    }
}

// ---------------- GEMM1: H1 = (X .* out_norm) @ W1   (nN x 128) @ (128 x 64)
// one wave -> 16x16 output tile via V_WMMA_F32_16X16X4_F32; 4 waves cover 16x64
__global__ __launch_bounds__(128) void gcn_gemm1_kernel(
        const float* __restrict__ X, const float* __restrict__ W,
        const float* __restrict__ onorm, float* __restrict__ H, int nN) {
    const int mt   = blockIdx.x;
    const int wv   = threadIdx.x >> 5;      // wave id == N tile (0..3)
    const int lane = threadIdx.x & 31;
    const int half = lane >> 4;             // 0: K pair {0,1}; 1: K pair {2,3}
    const int l    = lane & 15;
    const int m0   = mt * 16;
    const int n0   = wv * 16;

    int row = m0 + l;
    if (row >= nN) row = nN - 1;            // clamp loads, keep EXEC all-1
    const float  sc = onorm[row];
    const float* xr = X + (long long)row * IN_F;

    v8f acc = {};
#pragma unroll
    for (int k0 = 0; k0 < IN_F; k0 += 4) {
        const int ka = k0 + 2 * half;
        v2f a = *(const v2f*)(xr + ka);     // X[row, ka..ka+1]
        a *= sc;                            // fused out_norm scaling
        v2f b;
        b.x = W[(long long)ka       * HID_F + n0 + l];
        b.y = W[(long long)(ka + 1) * HID_F + n0 + l];
        acc = __builtin_amdgcn_wmma_f32_16x16x4_f32(
                  false, a, false, b, (short)0, acc, false, false);
    }
#pragma unroll
    for (int r = 0; r < 8; ++r) {
        const int orow = m0 + r + 8 * half;
        if (orow < nN) H[(long long)orow * HID_F + n0 + l] = acc[r];
    }
}

// ---------------- scatter 1: AGG[dst, c] += H[src, c] ----------------
__global__ void gcn_scatter1_kernel(const int* __restrict__ src,
                                    const int* __restrict__ dst,
                                    const float* __restrict__ H,
                                    float* __restrict__ AGG, long long total) {
    long long t = (long long)blockIdx.x * blockDim.x + threadIdx.x;
    if (t >= total) return;
    const int e = (int)(t >> 6);
    const int c = (int)(t & 63);
    const int s = src[e];
    const int d = dst[e];
    atomicAdd(&AGG[(long long)d * HID_F + c], H[(long long)s * HID_F + c]);
}

// h = relu(AGG .* in_norm + b1), in place
__global__ void gcn_post1_kernel(float* __restrict__ A,
                                 const float* __restrict__ inorm,
                                 const float* __restrict__ b1, long long n) {
    long long i = (long long)blockIdx.x * blockDim.x + threadIdx.x;
    if (i < n) {
        const int node = (int)(i >> 6);
        const int c    = (int)(i & 63);
        float v = A[i] * inorm[node] + b1[c];
        A[i] = v > 0.0f ? v : 0.0f;
    }
}

// ---------------- scatter 2 (fused src scaling): AGG[dst,c] += h[src,c]*onorm[src]
__global__ void gcn_scatter2_kernel(const int* __restrict__ src,
                                    const int* __restrict__ dst,
                                    const float* __restrict__ H,
                                    const float* __restrict__ onorm,
                                    float* __restrict__ AGG, long long total) {
    long long t = (long long)blockIdx.x * blockDim.x + threadIdx.x;
    if (t >= total) return;
    const int e = (int)(t >> 6);
    const int c = (int)(t & 63);
    const int s = src[e];
    const int d = dst[e];
    atomicAdd(&AGG[(long long)d * HID_F + c],
              H[(long long)s * HID_F + c] * onorm[s]);
}

// ---------------- GEMM2 + epilogue: out = (AGG @ W2) .* in_norm + b2
__global__ __launch_bounds__(128) void gcn_gemm2_kernel(
        const float* __restrict__ A, const float* __restrict__ W,
        const float* __restrict__ inorm, const float* __restrict__ bias,
        float* __restrict__ Out, int nN) {
    const int mt   = blockIdx.x;
    const int wv   = threadIdx.x >> 5;
    const int lane = threadIdx.x & 31;
    const int half = lane >> 4;
    const int l    = lane & 15;
    const int m0   = mt * 16;
    const int n0   = wv * 16;

    int row = m0 + l;
    if (row >= nN) row = nN - 1;
    const float* ar = A + (long long)row * HID_F;

    v8f acc = {};
#pragma unroll
    for (int k0 = 0; k0 < HID_F; k0 += 4) {
        const int ka = k0 + 2 * half;
        v2f a = *(const v2f*)(ar + ka);
        v2f b;
        b.x = W[(long long)ka       * OUT_F + n0 + l];
        b.y = W[(long long)(ka + 1) * OUT_F + n0 + l];
        acc = __builtin_amdgcn_wmma_f32_16x16x4_f32(
                  false, a, false, b, (short)0, acc, false, false);
    }
    const float bcol = bias[n0 + l];
#pragma unroll
    for (int r = 0; r < 8; ++r) {
        const int orow = m0 + r + 8 * half;
        if (orow < nN)
            Out[(long long)orow * OUT_F + n0 + l] = acc[r] * inorm[orow] + bcol;
    }
}

extern "C" void kernel_launch(void* const* d_in, const int* in_sizes, int n_in,
                              void* d_out, int out_size, void* d_ws, size_t ws_size,
                              hipStream_t stream) {
    const float* X   = (const float*)d_in[0];
    const int*   src = (const int*)  d_in[1];
    const int*   dst = (const int*)  d_in[2];
    const float* W1  = (const float*)d_in[3];
    const float* b1  = (const float*)d_in[4];
    const float* W2  = (const float*)d_in[5];
    const float* b2  = (const float*)d_in[6];
    float*       out = (float*)d_out;

    const int nN = in_sizes[0] / IN_F;   // 100000
    const int nE = in_sizes[1];          // 1600000

    // workspace layout (floats): [out_norm nN][in_norm nN][bufA nN*64][bufB nN*64]
    float* ws    = (float*)d_ws;
    float* onorm = ws;
    float* inorm = ws + nN;
    float* bufA  = ws + 2LL * nN;                 // H1, later AGG2
    float* bufB  = bufA + (long long)nN * HID_F;  // AGG1, later h

    const long long nh  = (long long)nN * HID_F;  // 6.4M
    const long long tot = (long long)nE * HID_F;  // 102.4M (edge,col) pairs

    // degrees -> norms (accumulate into norm buffers, finalize in place)
    gcn_zero_kernel  <<<(2LL*nN + 255) / 256, 256, 0, stream>>>(ws, 2LL*nN);
    gcn_degree_kernel<<<(nE + 255) / 256, 256, 0, stream>>>(src, dst, onorm, inorm, nE);
    gcn_norm_kernel  <<<(2LL*nN + 255) / 256, 256, 0, stream>>>(ws, 2LL*nN);

    // layer 1
    gcn_gemm1_kernel   <<<(nN + 15) / 16, 128, 0, stream>>>(X, W1, onorm, bufA, nN);
    gcn_zero_kernel    <<<(nh + 255) / 256, 256, 0, stream>>>(bufB, nh);
    gcn_scatter1_kernel<<<(tot + 255) / 256, 256, 0, stream>>>(src, dst, bufA, bufB, tot);
    gcn_post1_kernel   <<<(nh + 255) / 256, 256, 0, stream>>>(bufB, inorm, b1, nh);

    // layer 2 (out_norm scaling fused into gather)
    gcn_zero_kernel    <<<(nh + 255) / 256, 256, 0, stream>>>(bufA, nh);
    gcn_scatter2_kernel<<<(tot + 255) / 256, 256, 0, stream>>>(src, dst, bufB, onorm, bufA, tot);
    gcn_gemm2_kernel   <<<(nN + 15) / 16, 128, 0, stream>>>(bufA, W2, inorm, b2, out, nN);
}